// GAT_76699525972342
// MI455X (gfx1250) — compile-verified
//
#include <hip/hip_runtime.h>
#include <cstdint>

// ---------------------------------------------------------------------------
// GAT 2-layer encode for MI455X (gfx1250, wave32).
// GEMMs via V_WMMA_F32_16X16X4_F32 (fp32 in/out, 16x16 tile per wave).
// Edge softmax/scatter via L2-resident atomics (192MB L2 holds all node state).
// ---------------------------------------------------------------------------

typedef float v2f __attribute__((ext_vector_type(2)));
typedef float v8f __attribute__((ext_vector_type(8)));

#define NEG_SLOPE 0.2f
#define GAT_EPS 1e-16f
// ordered-uint encoding of -inf (see enc_ord)
#define ENC_NEG_INF 0x007FFFFFu

__device__ __forceinline__ unsigned enc_ord(float f) {
  unsigned u = __float_as_uint(f);
  return (u & 0x80000000u) ? ~u : (u | 0x80000000u);
}
__device__ __forceinline__ float dec_ord(unsigned k) {
  return (k & 0x80000000u) ? __uint_as_float(k & 0x7FFFFFFFu)
                           : __uint_as_float(~k);
}

// ---------------------------------------------------------------------------
// Generic fill (grid-stride)
// ---------------------------------------------------------------------------
__global__ void fill_u32_kernel(unsigned* __restrict__ p, unsigned v, long long n) {
  long long i = (long long)blockIdx.x * blockDim.x + threadIdx.x;
  long long stride = (long long)gridDim.x * blockDim.x;
  for (; i < n; i += stride) p[i] = v;
}

// ---------------------------------------------------------------------------
// fp32 GEMM: C[M,N] = A[M,K] @ B[K,N] via V_WMMA_F32_16X16X4_F32.
// One wave32 per 16x16 output tile. M,N multiples of 16; K multiple of 4.
// A fragment (16x4 f32): lane = (k>=2)*16 + m, vgpr = k%2.
// B fragment (4x16 f32): lane = (k>=2)*16 + n, vgpr = k%2.
// C fragment (16x16 f32): vgpr r -> row r (lanes 0-15) / row r+8 (lanes 16-31).
// ---------------------------------------------------------------------------
__global__ void gemm_f32_wmma_kernel(const float* __restrict__ A,
                                     const float* __restrict__ B,
                                     float* __restrict__ C,
                                     int M, int N, int K) {
  int wave = (int)((blockIdx.x * blockDim.x + threadIdx.x) >> 5);
  int lane = threadIdx.x & 31;
  int ntiles = N >> 4;
  int mtiles = (M + 15) >> 4;
  if (wave >= mtiles * ntiles) return;   // wave-uniform: EXEC stays all-ones
  int mt = wave / ntiles;
  int nt = wave - mt * ntiles;
  int m0 = mt << 4, n0 = nt << 4;
  int half = lane >> 4;       // 0: K={0,1}, 1: K={2,3}
  int lid  = lane & 15;

  v8f c = {0.f, 0.f, 0.f, 0.f, 0.f, 0.f, 0.f, 0.f};
  const float* arow = A + (size_t)(m0 + lid) * K + 2 * half;
  const float* bcol = B + (size_t)(2 * half) * N + n0 + lid;

  for (int k0 = 0; k0 < K; k0 += 4) {
    v2f a, b;
    a.x = arow[k0];
    a.y = arow[k0 + 1];
    b.x = bcol[(size_t)k0 * N];
    b.y = bcol[(size_t)k0 * N + N];
    c = __builtin_amdgcn_wmma_f32_16x16x4_f32(false, a, false, b,
                                              (short)0, c, false, false);
  }

  float* crow = C + (size_t)m0 * N + n0 + lid;
#pragma unroll
  for (int r = 0; r < 8; ++r)
    crow[(size_t)(r + 8 * half) * N] = c[r];
}

// ---------------------------------------------------------------------------
// Per-node attention logits: as[n,h] = sum_d h[n,h,d] * a_src[h,d] (ditto dst)
// ---------------------------------------------------------------------------
template <int H, int D>
__global__ void node_alpha_kernel(const float* __restrict__ h,
                                  const float* __restrict__ avs,
                                  const float* __restrict__ avd,
                                  float* __restrict__ as,
                                  float* __restrict__ ad, int N) {
  int n = blockIdx.x * blockDim.x + threadIdx.x;
  if (n >= N) return;
  const float* hr = h + (size_t)n * H * D;
#pragma unroll
  for (int hh = 0; hh < H; ++hh) {
    float s0 = 0.f, s1 = 0.f;
#pragma unroll
    for (int d = 0; d < D; ++d) {
      float v = hr[hh * D + d];
      s0 += v * avs[hh * D + d];
      s1 += v * avd[hh * D + d];
    }
    as[(size_t)n * H + hh] = s0;
    ad[(size_t)n * H + hh] = s1;
  }
}

// ---------------------------------------------------------------------------
// Edge pass 1: segment max of leaky(e) into m[dst,h] (ordered-uint atomicMax)
// One thread per (edge, head). Edges idx>=E are virtual self-loops.
// ---------------------------------------------------------------------------
template <int H>
__global__ void edge_max_kernel(const int* __restrict__ src,
                                const int* __restrict__ dst, int E, int Etot,
                                const float* __restrict__ as,
                                const float* __restrict__ ad,
                                unsigned* __restrict__ m) {
  int gid = blockIdx.x * blockDim.x + threadIdx.x;
  if (gid >= Etot * H) return;
  int eidx = gid / H, hh = gid - (gid / H) * H;
  int si, di;
  if (eidx < E) { si = src[eidx]; di = dst[eidx]; }
  else          { si = di = eidx - E; }
  float e = as[(size_t)si * H + hh] + ad[(size_t)di * H + hh];
  e = e > 0.f ? e : NEG_SLOPE * e;
  atomicMax(&m[(size_t)di * H + hh], enc_ord(e));
}

// ---------------------------------------------------------------------------
// Edge pass 2: s[dst,h] += exp(leaky(e) - m[dst,h])
// ---------------------------------------------------------------------------
template <int H>
__global__ void edge_sum_kernel(const int* __restrict__ src,
                                const int* __restrict__ dst, int E, int Etot,
                                const float* __restrict__ as,
                                const float* __restrict__ ad,
                                const unsigned* __restrict__ m,
                                float* __restrict__ s) {
  int gid = blockIdx.x * blockDim.x + threadIdx.x;
  if (gid >= Etot * H) return;
  int eidx = gid / H, hh = gid - (gid / H) * H;
  int si, di;
  if (eidx < E) { si = src[eidx]; di = dst[eidx]; }
  else          { si = di = eidx - E; }
  float e = as[(size_t)si * H + hh] + ad[(size_t)di * H + hh];
  e = e > 0.f ? e : NEG_SLOPE * e;
  float ex = __expf(e - dec_ord(m[(size_t)di * H + hh]));
  atomicAdd(&s[(size_t)di * H + hh], ex);
}

// ---------------------------------------------------------------------------
// Edge pass 3: acc[dst, :] += h[src, :] * alpha. One wave32 per edge;
// lanes cover contiguous channels -> coalesced gathers + cacheline-merged
// atomics in L2.
// ---------------------------------------------------------------------------
template <int DTOT, int H>
__global__ void edge_scatter_kernel(const int* __restrict__ src,
                                    const int* __restrict__ dst, int E, int Etot,
                                    const float* __restrict__ as,
                                    const float* __restrict__ ad,
                                    const unsigned* __restrict__ m,
                                    const float* __restrict__ s,
                                    const float* __restrict__ hsrc,
                                    float* __restrict__ acc) {
  int gid = blockIdx.x * blockDim.x + threadIdx.x;
  int eidx = gid >> 5;
  int lane = gid & 31;
  if (eidx >= Etot) return;
  int si, di;
  if (eidx < E) { si = src[eidx]; di = dst[eidx]; }
  else          { si = di = eidx - E; }
  const int Dh = DTOT / H;
#pragma unroll
  for (int c = lane; c < DTOT; c += 32) {
    int hh = c / Dh;
    float e = as[(size_t)si * H + hh] + ad[(size_t)di * H + hh];
    e = e > 0.f ? e : NEG_SLOPE * e;
    float mm = dec_ord(m[(size_t)di * H + hh]);
    float ss = s[(size_t)di * H + hh];
    float alpha = __expf(e - mm) / (ss + GAT_EPS);
    atomicAdd(&acc[(size_t)di * DTOT + c],
              hsrc[(size_t)si * DTOT + c] * alpha);
  }
}

// ---------------------------------------------------------------------------
// Epilogue: acc += bias (per channel), optional ELU
// ---------------------------------------------------------------------------
__global__ void finalize_kernel(float* __restrict__ acc,
                                const float* __restrict__ bias,
                                long long total, int C, int apply_elu) {
  long long i = (long long)blockIdx.x * blockDim.x + threadIdx.x;
  if (i >= total) return;
  int j = (int)(i % C);
  float v = acc[i] + bias[j];
  if (apply_elu) v = v > 0.f ? v : __expf(v) - 1.f;
  acc[i] = v;
}

// ---------------------------------------------------------------------------
static inline int cdiv(long long a, long long b) { return (int)((a + b - 1) / b); }

extern "C" void kernel_launch(void* const* d_in, const int* in_sizes, int n_in,
                              void* d_out, int out_size, void* d_ws, size_t ws_size,
                              hipStream_t stream) {
  const float* x      = (const float*)d_in[0];
  const int*   ei     = (const int*)  d_in[1];   // int32 (JAX canonicalized)
  const float* W1     = (const float*)d_in[2];
  const float* a_src1 = (const float*)d_in[3];
  const float* a_dst1 = (const float*)d_in[4];
  const float* b1     = (const float*)d_in[5];
  const float* W2     = (const float*)d_in[6];
  const float* a_src2 = (const float*)d_in[7];
  const float* a_dst2 = (const float*)d_in[8];
  const float* b2     = (const float*)d_in[9];
  float* z = (float*)d_out;

  const int Fin = 256, H1 = 8, C1 = 64, C2 = 128;
  const int N    = in_sizes[0] / Fin;     // 100000
  const int E    = in_sizes[1] / 2;       // 1600000
  const int Etot = E + N;                 // + self loops
  const int* srcv = ei;
  const int* dstv = ei + E;

  // ---- workspace layout (floats) ----
  float*    h1   = (float*)d_ws;                      // N*64
  float*    as1  = h1  + (size_t)N * C1;              // N*8
  float*    ad1  = as1 + (size_t)N * H1;              // N*8
  unsigned* m1   = (unsigned*)(ad1 + (size_t)N * H1); // N*8
  float*    s1   = (float*)(m1 + (size_t)N * H1);     // N*8
  float*    acc1 = s1  + (size_t)N * H1;              // N*64 (becomes ELU'd hidden)
  float*    h2   = acc1 + (size_t)N * C1;             // N*128
  float*    as2  = h2  + (size_t)N * C2;              // N
  float*    ad2  = as2 + N;                           // N
  unsigned* m2   = (unsigned*)(ad2 + N);              // N
  float*    s2   = (float*)(m2 + N);                  // N

  const int B = 256;

  // ---- init accumulators (s1 and acc1 are contiguous) ----
  fill_u32_kernel<<<2048, B, 0, stream>>>((unsigned*)s1, 0u,
                                          (long long)N * (H1 + C1));
  fill_u32_kernel<<<2048, B, 0, stream>>>(m1, ENC_NEG_INF, (long long)N * H1);
  fill_u32_kernel<<<2048, B, 0, stream>>>((unsigned*)z, 0u, (long long)N * C2);
  fill_u32_kernel<<<2048, B, 0, stream>>>((unsigned*)s2, 0u, (long long)N);
  fill_u32_kernel<<<2048, B, 0, stream>>>(m2, ENC_NEG_INF, (long long)N);

  // ---- layer 1 ----
  {
    int waves = (N / 16) * (C1 / 16);  // 25000
    gemm_f32_wmma_kernel<<<cdiv((long long)waves * 32, B), B, 0, stream>>>(
        x, W1, h1, N, C1, Fin);
  }
  node_alpha_kernel<8, 8><<<cdiv(N, B), B, 0, stream>>>(h1, a_src1, a_dst1,
                                                        as1, ad1, N);
  edge_max_kernel<8><<<cdiv((long long)Etot * 8, B), B, 0, stream>>>(
      srcv, dstv, E, Etot, as1, ad1, m1);
  edge_sum_kernel<8><<<cdiv((long long)Etot * 8, B), B, 0, stream>>>(
      srcv, dstv, E, Etot, as1, ad1, m1, s1);
  edge_scatter_kernel<64, 8><<<cdiv((long long)Etot * 32, B), B, 0, stream>>>(
      srcv, dstv, E, Etot, as1, ad1, m1, s1, h1, acc1);
  finalize_kernel<<<cdiv((long long)N * C1, B), B, 0, stream>>>(
      acc1, b1, (long long)N * C1, C1, /*elu=*/1);

  // ---- layer 2 ----
  {
    int waves = (N / 16) * (C2 / 16);  // 50000
    gemm_f32_wmma_kernel<<<cdiv((long long)waves * 32, B), B, 0, stream>>>(
        acc1, W2, h2, N, C2, C1);
  }
  node_alpha_kernel<1, 128><<<cdiv(N, B), B, 0, stream>>>(h2, a_src2, a_dst2,
                                                          as2, ad2, N);
  edge_max_kernel<1><<<cdiv((long long)Etot, B), B, 0, stream>>>(
      srcv, dstv, E, Etot, as2, ad2, m2);
  edge_sum_kernel<1><<<cdiv((long long)Etot, B), B, 0, stream>>>(
      srcv, dstv, E, Etot, as2, ad2, m2, s2);
  edge_scatter_kernel<128, 1><<<cdiv((long long)Etot * 32, B), B, 0, stream>>>(
      srcv, dstv, E, Etot, as2, ad2, m2, s2, h2, z);
  finalize_kernel<<<cdiv((long long)N * C2, B), B, 0, stream>>>(
      z, b2, (long long)N * C2, C2, /*elu=*/0);
}